// LightGCNStack_36197984371493
// MI455X (gfx1250) — compile-verified
//
#include <hip/hip_runtime.h>

// LightGCN bipartite propagation for MI455X (gfx1250, wave32).
// Memory/atomics-bound: total traffic ~6.5 GB, working set ~130 MB -> fully
// L2-resident (192 MB). Scatter = float4 gathers + global_atomic_add_f32 into
// L2. Degree normalization fused via V_WMMA_F32_16X16X4_F32 diagonal multiply.

#define LD 64           // LATENT_DIM
#define NLAYERS 3

typedef __attribute__((ext_vector_type(2))) float v2f;
typedef __attribute__((ext_vector_type(8))) float v8f;

// ---------------------------------------------------------------- zero fill
__global__ void zero_f4(float4* __restrict__ p, long n4) {
    long i = (long)blockIdx.x * blockDim.x + threadIdx.x;
    if (i < n4) p[i] = make_float4(0.f, 0.f, 0.f, 0.f);
}

// ------------------------------------------------------------ degree counts
__global__ void count_kernel(const int* __restrict__ art,
                             const int* __restrict__ usr,
                             float* __restrict__ cnt_a,
                             float* __restrict__ cnt_u, int E) {
    int e = blockIdx.x * blockDim.x + threadIdx.x;
    if (e < E) {
        atomicAdd(cnt_a + art[e], 1.0f);
        atomicAdd(cnt_u + usr[e], 1.0f);
    }
}

__global__ void inv_kernel(float* __restrict__ c, int n) {
    int i = blockIdx.x * blockDim.x + threadIdx.x;
    if (i < n) c[i] = 1.0f / fmaxf(c[i], 1.0f);
}

// --------------------------------------------------- final_* = x * scale
__global__ void init_scale(const float4* __restrict__ x,
                           float4* __restrict__ out, float s, long n4) {
    long i = (long)blockIdx.x * blockDim.x + threadIdx.x;
    if (i < n4) {
        float4 v = x[i];
        out[i] = make_float4(v.x * s, v.y * s, v.z * s, v.w * s);
    }
}

// ------------------------------------------------------------ edge scatter
// 16 lanes per edge; each lane moves one float4 of the 64-float row.
// Gathers are 256B-coalesced per edge; sums accumulate via L2 f32 atomics.
__global__ void scatter_kernel(const float4* __restrict__ src,
                               const int* __restrict__ srcIds,
                               const int* __restrict__ dstIds,
                               float* __restrict__ dst, int E) {
    long tid = (long)blockIdx.x * blockDim.x + threadIdx.x;
    long total = (long)E * 16;
    if (tid >= total) return;
    int e = (int)(tid >> 4);
    int q = (int)(tid & 15);
    int sr = srcIds[e];
    int dr = dstIds[e];
    float4 v = src[(long)sr * 16 + q];            // global_load_b128
    float* d = dst + (long)dr * LD + q * 4;
    atomicAdd(d + 0, v.x);                        // global_atomic_add_f32 x4
    atomicAdd(d + 1, v.y);
    atomicAdd(d + 2, v.z);
    atomicAdd(d + 3, v.w);
}

// ---------------------------------------------- finalize: X = diag(inv) * X,
// fin += scale * X. One wave32 per 16-row block; 4 column groups of 16;
// each 16x16 tile = 4 chained V_WMMA_F32_16X16X4_F32 with diagonal A chunks.
// Diagonal product => exactly one nonzero multiply per output element, so the
// WMMA result equals the scalar multiply bit-for-bit.
__global__ void finalize_wmma(float* __restrict__ X,
                              const float* __restrict__ inv,
                              float* __restrict__ fin,
                              float scale, int nRowBlocks) {
    int gtid = blockIdx.x * blockDim.x + threadIdx.x;
    int wave = gtid >> 5;          // wave32
    int lane = gtid & 31;
    if (wave >= nRowBlocks) return;   // wave-uniform: EXEC stays all-ones

    int m       = lane & 15;
    int rowBase = wave * 16;
    float iv    = inv[rowBase + m];
    int kl0     = (lane < 16) ? 0 : 2;   // A/B K-slot held by this lane half
    int mBase   = (lane < 16) ? 0 : 8;   // C/D row base for this lane half

#pragma unroll
    for (int ng = 0; ng < 4; ++ng) {      // 4 groups of 16 columns (LD=64)
        int colBase = ng * 16;
        v8f c = {};
#pragma unroll
        for (int kc = 0; kc < 4; ++kc) {  // chain K: 16 = 4 x (16x16x4)
            // A chunk: columns kc*4..kc*4+3 of diag(inv).
            // A layout (16x4 f32): lanes 0-15 M=0..15 {K0,K1}, lanes 16-31 {K2,K3}.
            v2f a; a.x = 0.f; a.y = 0.f;
            if (m == kc * 4 + kl0)     a.x = iv;
            if (m == kc * 4 + kl0 + 1) a.y = iv;
            // B chunk: rows kc*4..kc*4+3 of X, cols colBase..colBase+15.
            // B layout (4x16 f32): lanes 0-15 N=0..15 {K0,K1}, lanes 16-31 {K2,K3}.
            const float* bp = X + (long)(rowBase + kc * 4 + kl0) * LD + colBase + m;
            v2f b; b.x = bp[0]; b.y = bp[LD];
            c = __builtin_amdgcn_wmma_f32_16x16x4_f32(
                    false, a, false, b, (short)0, c, false, false);
        }
        // D layout: VGPR j -> (M = mBase + j, N = m) for this lane half.
#pragma unroll
        for (int j = 0; j < 8; ++j) {
            long off = (long)(rowBase + mBase + j) * LD + colBase + m;
            float v  = c[j];
            X[off]   = v;                 // x_new, in place (tile-disjoint)
            fin[off] += scale * v;        // running LightGCN average
        }
    }
}

// ----------------------------------------------------------------- launcher
extern "C" void kernel_launch(void* const* d_in, const int* in_sizes, int n_in,
                              void* d_out, int out_size, void* d_ws, size_t ws_size,
                              hipStream_t stream) {
    const float* xu = (const float*)d_in[0];           // [NU, 64]
    const float* xa = (const float*)d_in[1];           // [NA, 64]
    const int*   e0 = (const int*)d_in[2];             // a2u: [2, E] = {art, usr}
    const int E  = in_sizes[2] / 2;
    const int NU = in_sizes[0] / LD;
    const int NA = in_sizes[1] / LD;
    const int* art = e0;        // a2u row 0: artist (source) ids
    const int* usr = e0 + E;    // a2u row 1: user (target) ids

    // Workspace layout (floats): buf_u | buf_a | cnt_u | cnt_a  (~65 MB)
    float* buf_u = (float*)d_ws;
    float* buf_a = buf_u + (size_t)NU * LD;
    float* cnt_u = buf_a + (size_t)NA * LD;
    float* cnt_a = cnt_u + NU;
    float* fin_u = (float*)d_out;                      // outputs concatenated
    float* fin_a = fin_u + (size_t)NU * LD;

    const float scale = 1.0f / (NLAYERS + 1);
    const int BLK = 256;                               // 8 wave32s per block
    auto blocks = [](long n, int b) { return (unsigned)((n + b - 1) / b); };

    // 1) degree counts (layer-invariant) -> reciprocal in place
    {
        long ncf4 = (long)(NU + NA) / 4;               // cnt_u,cnt_a contiguous
        zero_f4<<<blocks(ncf4, BLK), BLK, 0, stream>>>((float4*)cnt_u, ncf4);
        count_kernel<<<blocks(E, BLK), BLK, 0, stream>>>(art, usr, cnt_a, cnt_u, E);
        inv_kernel<<<blocks(NU + NA, BLK), BLK, 0, stream>>>(cnt_u, NU + NA);
    }

    // 2) final_* = x_* * scale
    init_scale<<<blocks((long)NU * LD / 4, BLK), BLK, 0, stream>>>(
        (const float4*)xu, (float4*)fin_u, scale, (long)NU * LD / 4);
    init_scale<<<blocks((long)NA * LD / 4, BLK), BLK, 0, stream>>>(
        (const float4*)xa, (float4*)fin_a, scale, (long)NA * LD / 4);

    // 3) three propagation layers, ping-ponging through buf_u / buf_a
    const long scatThreads = (long)E * 16;
    const float* srcArtists = xa;                      // layer-0 artist features
    for (int layer = 0; layer < NLAYERS; ++layer) {
        // artists -> users
        long nu4 = (long)NU * LD / 4;
        zero_f4<<<blocks(nu4, BLK), BLK, 0, stream>>>((float4*)buf_u, nu4);
        scatter_kernel<<<blocks(scatThreads, BLK), BLK, 0, stream>>>(
            (const float4*)srcArtists, art, usr, buf_u, E);
        finalize_wmma<<<blocks((long)(NU / 16) * 32, BLK), BLK, 0, stream>>>(
            buf_u, cnt_u, fin_u, scale, NU / 16);

        // users -> artists
        long na4 = (long)NA * LD / 4;
        zero_f4<<<blocks(na4, BLK), BLK, 0, stream>>>((float4*)buf_a, na4);
        scatter_kernel<<<blocks(scatThreads, BLK), BLK, 0, stream>>>(
            (const float4*)buf_u, usr, art, buf_a, E);
        finalize_wmma<<<blocks((long)(NA / 16) * 32, BLK), BLK, 0, stream>>>(
            buf_a, cnt_a, fin_a, scale, NA / 16);

        srcArtists = buf_a;                            // next layer's input
    }
}